// GELU264_23648089932059
// MI455X (gfx1250) — compile-verified
//
#include <hip/hip_runtime.h>
#include <math.h>

// ---------------------------------------------------------------------------
// Sizes fixed by the reference: B*T = 16384 rows, D = 1024, N = 512.
// Pass1/2 assume D == 4*TPB (=1024) and N <= 512; dims are still taken from
// in_sizes so the scalar plumbing stays generic.
// ---------------------------------------------------------------------------
#define TPB   256
#define NB1   256
#define D_MAX 1024
#define N_MAX 512
#define EPSF  1e-12f

typedef __attribute__((ext_vector_type(2))) float v2f;
typedef __attribute__((ext_vector_type(4))) float v4f;   // native vector: OK for NT builtins
typedef __attribute__((ext_vector_type(8))) float v8f;

__device__ __forceinline__ float gelu_f(float x) {
    const float c = 0.7978845608028654f; // sqrt(2/pi)
    float x3 = x * x * x;
    float t = tanhf(c * (x + 0.044715f * x3));
    return 0.5f * x * (1.0f + t);
}

// ---------------------------------------------------------------------------
// Pass 1: y = gelu(x); per-block column partial sums -> partials[NB1][D]
// ---------------------------------------------------------------------------
__global__ void __launch_bounds__(TPB)
pass1_gelu_colsum(const float* __restrict__ x, float* __restrict__ y,
                  float* __restrict__ partials, int rows, int D) {
    const int t = threadIdx.x;          // thread owns columns 4t..4t+3
    float a0 = 0.f, a1 = 0.f, a2 = 0.f, a3 = 0.f;
    for (int r = blockIdx.x; r < rows; r += NB1) {
        const v4f* xr = (const v4f*)(x + (size_t)r * D);
        v4f v = __builtin_nontemporal_load(&xr[t]);  // x is read exactly once
        v4f g;
        g.x = gelu_f(v.x); g.y = gelu_f(v.y);
        g.z = gelu_f(v.z); g.w = gelu_f(v.w);
        v4f* yr = (v4f*)(y + (size_t)r * D);
        yr[t] = g;                                   // RT: let y linger in L2
        a0 += g.x; a1 += g.y; a2 += g.z; a3 += g.w;
    }
    v4f* p = (v4f*)(partials + (size_t)blockIdx.x * D);
    v4f acc; acc.x = a0; acc.y = a1; acc.z = a2; acc.w = a3;
    p[t] = acc;
}

// ---------------------------------------------------------------------------
// Pass 2: single workgroup. Reduce partials -> mean -> m_n; sim_g; buf row
// norms; sims via V_WMMA_F32_16X16X4_F32; argmax; scalar gate -> gate_out[0].
// ---------------------------------------------------------------------------
__global__ void __launch_bounds__(TPB)
pass2_gate(const float* __restrict__ partials,
           const float* __restrict__ buf,
           const float* __restrict__ global_mean,
           const float* __restrict__ facil_l,
           const unsigned char* __restrict__ mask,
           const float* __restrict__ log_k_local,
           const float* __restrict__ log_k_global,
           const float* __restrict__ facil_g,
           float* __restrict__ gate_out,
           int rows, int D, int N) {
    __shared__ float mn[D_MAX];     // normalized mean vector
    __shared__ float rn[N_MAX];     // 1 / ||buf_row||
    __shared__ float sims[N_MAX];   // masked similarities
    __shared__ float redf[TPB];
    __shared__ int   redi[TPB];

    const int t = threadIdx.x;

    // ---- column sums of partials (fixed order => deterministic) ----
    float s0 = 0.f, s1 = 0.f, s2 = 0.f, s3 = 0.f;
    for (int b = 0; b < NB1; ++b) {
        v4f v = ((const v4f*)(partials + (size_t)b * D))[t];
        s0 += v.x; s1 += v.y; s2 += v.z; s3 += v.w;
    }
    const float invR = 1.0f / (float)rows;
    float m0 = s0 * invR, m1 = s1 * invR, m2 = s2 * invR, m3 = s3 * invR;

    // ---- ||mean|| via block tree reduce ----
    redf[t] = m0 * m0 + m1 * m1 + m2 * m2 + m3 * m3;
    __syncthreads();
    for (int s = TPB / 2; s > 0; s >>= 1) {
        if (t < s) redf[t] += redf[t + s];
        __syncthreads();
    }
    const float nrm = fmaxf(sqrtf(redf[0]), EPSF);
    __syncthreads();
    const float innrm = 1.0f / nrm;
    mn[4 * t + 0] = m0 * innrm;
    mn[4 * t + 1] = m1 * innrm;
    mn[4 * t + 2] = m2 * innrm;
    mn[4 * t + 3] = m3 * innrm;

    // ---- sim_g = dot(m_n, global_mean) ----
    redf[t] = mn[4 * t + 0] * global_mean[4 * t + 0]
            + mn[4 * t + 1] * global_mean[4 * t + 1]
            + mn[4 * t + 2] * global_mean[4 * t + 2]
            + mn[4 * t + 3] * global_mean[4 * t + 3];
    __syncthreads();
    for (int s = TPB / 2; s > 0; s >>= 1) {
        if (t < s) redf[t] += redf[t + s];
        __syncthreads();
    }
    const float sim_g_raw = redf[0];
    __syncthreads();

    // ---- buf row inverse norms ----
    for (int row = t; row < N; row += TPB) {
        const float* r = buf + (size_t)row * D;
        float acc = 0.f;
        for (int k = 0; k < D; k += 4) {
            v4f v = *(const v4f*)(r + k);
            acc += v.x * v.x + v.y * v.y + v.z * v.z + v.w * v.w;
        }
        rn[row] = 1.0f / fmaxf(sqrtf(acc), EPSF);
    }
    __syncthreads();

    // ---- sims = (buf @ m_n) * rn  via f32 WMMA 16x16x4 -------------------
    // A (16x4 f32): lanes 0-15 hold {K=0,K=1}, lanes 16-31 hold {K=2,K=3},
    // row M = lane&15. B (4x16) mirrored; m_n replicated across all 16 cols,
    // so every result column equals the row dot product.
    const int wave = t >> 5;
    const int lane = t & 31;
    const bool lo  = lane < 16;
    const int rsel = lane & 15;
    const int koff = lo ? 0 : 2;
    const int nTiles = N >> 4;
    for (int tile = wave; tile < nTiles; tile += (TPB >> 5)) {
        const int rowA = tile * 16 + rsel;
        const float* arow = buf + (size_t)rowA * D;
        v8f c = {0.f, 0.f, 0.f, 0.f, 0.f, 0.f, 0.f, 0.f};
        for (int k = 0; k < D; k += 4) {
            v2f a, b;
            a.x = arow[k + koff];
            a.y = arow[k + koff + 1];
            b.x = mn[k + koff];
            b.y = mn[k + koff + 1];
            c = __builtin_amdgcn_wmma_f32_16x16x4_f32(
                    /*neg_a=*/false, a, /*neg_b=*/false, b,
                    /*c_mod=*/(short)0, c, /*reuse_a=*/false, /*reuse_b=*/false);
        }
        // C/D layout: VGPR v, lanes 0-15 -> M = v, lanes 16-31 -> M = v+8.
        if (rsel == 0) {
            const int base = tile * 16 + (lo ? 0 : 8);
            #pragma unroll
            for (int v = 0; v < 8; ++v) {
                const int row = base + v;
                const float sv = c[v] * rn[row];
                sims[row] = mask[row] ? sv : -1.0f;
            }
        }
    }
    __syncthreads();

    // ---- argmax (first index wins ties, matching jnp.argmax) ----
    float best = -INFINITY; int bidx = 0;
    for (int row = t; row < N; row += TPB) {
        const float v = sims[row];
        if (v > best || (v == best && row < bidx)) { best = v; bidx = row; }
    }
    redf[t] = best; redi[t] = bidx;
    __syncthreads();
    for (int s = TPB / 2; s > 0; s >>= 1) {
        if (t < s) {
            const float vo = redf[t + s];
            const int  io = redi[t + s];
            if (vo > redf[t] || (vo == redf[t] && io < redi[t])) {
                redf[t] = vo; redi[t] = io;
            }
        }
        __syncthreads();
    }

    if (t == 0) {
        const int nearest = redi[0];
        const float sim_l = fminf(fmaxf(redf[0], 0.f), 1.f);
        const float sim_gc = fminf(fmaxf(sim_g_raw, 0.f), 1.f);
        const bool fire = sim_l > 0.85f;
        const float fl = facil_l[nearest] * (fire ? 2.0f : 1.0f);
        float fg = facil_g[0];
        if (fire) fg = fminf(fg * 1.5f, 16.0f);
        const float kl = fminf(fmaxf(expf(log_k_local[0]), 0.01f), 4.0f);
        const float kg = fminf(fmaxf(expf(log_k_global[0]), 0.01f), 4.0f);
        const float gate =
            fminf(1.0f + kl * (fl - 1.0f) * sim_l + kg * (fg - 1.0f) * sim_gc, 8.0f);
        gate_out[0] = gate;
    }
}

// ---------------------------------------------------------------------------
// Pass 3: out *= gate (y should still be resident in 192MB L2)
// ---------------------------------------------------------------------------
__global__ void __launch_bounds__(TPB)
pass3_scale(float* __restrict__ y, const float* __restrict__ gate, size_t n4) {
    const size_t i = (size_t)blockIdx.x * TPB + threadIdx.x;
    if (i >= n4) return;
    const float g = gate[0];
    v4f* p = (v4f*)y;
    v4f v = p[i];
    v.x *= g; v.y *= g; v.z *= g; v.w *= g;
    __builtin_nontemporal_store(v, &p[i]);   // final write, don't keep in cache
}

// ---------------------------------------------------------------------------
extern "C" void kernel_launch(void* const* d_in, const int* in_sizes, int n_in,
                              void* d_out, int out_size, void* d_ws, size_t ws_size,
                              hipStream_t stream) {
    const float*         x            = (const float*)d_in[0];
    const float*         log_k_local  = (const float*)d_in[1];
    const float*         log_k_global = (const float*)d_in[2];
    const float*         buf          = (const float*)d_in[3];
    const float*         facil_l      = (const float*)d_in[4];
    const float*         global_mean  = (const float*)d_in[5];
    const float*         facil_g      = (const float*)d_in[6];
    const unsigned char* mask         = (const unsigned char*)d_in[7];

    const int D    = in_sizes[5];          // 1024
    const int N    = in_sizes[4];          // 512
    const long tot = (long)in_sizes[0];    // B*T*D
    const int rows = (int)(tot / D);       // 16384

    float* y        = (float*)d_out;
    float* partials = (float*)d_ws;                    // NB1 * D floats (1 MB)
    float* gate     = partials + (size_t)NB1 * D;      // 1 float

    pass1_gelu_colsum<<<NB1, TPB, 0, stream>>>(x, y, partials, rows, D);
    pass2_gate<<<1, TPB, 0, stream>>>(partials, buf, global_mean, facil_l, mask,
                                      log_k_local, log_k_global, facil_g, gate,
                                      rows, D, N);
    const size_t n4 = (size_t)tot / 4;
    const int nb3 = (int)((n4 + TPB - 1) / TPB);
    pass3_scale<<<nb3, TPB, 0, stream>>>(y, gate, n4);
}